// CTCLayer_1133871366491
// MI455X (gfx1250) — compile-verified
//
#include <hip/hip_runtime.h>

// Problem constants (match reference)
#define Bsz    256
#define Tt     512
#define Cc     256
#define Uu     64
#define Ss     129        // 2*U+1
#define BLANKC 255        // blank = C-1
#define NEGV   (-1e30f)
#define EPSV   (1e-7f)
#define WIN    16         // rows per TDM window (16 rows * 1KB = 16KB tile)
#define NW     (Tt / WIN) // 32 windows

typedef unsigned int v4u __attribute__((ext_vector_type(4)));
typedef int          v4i __attribute__((ext_vector_type(4)));
typedef int          v8i __attribute__((ext_vector_type(8)));

// logaddexp of three terms: m + log(sum exp(x_i - m)).
// All-NEG case stays ~NEG (no NaN/Inf: -1e30 is finite, exp(huge neg)=0).
__device__ __forceinline__ float lae3(float x, float y, float z) {
    float m = fmaxf(x, fmaxf(y, z));
    float s = __expf(x - m) + __expf(y - m) + __expf(z - m);
    return m + __logf(s);
}
__device__ __forceinline__ float lae2(float x, float y) {
    float m = fmaxf(x, y);
    return m + __logf(__expf(x - m) + __expf(y - m));
}

// Issue one TDM 2-D tile load: WIN rows x Cc floats from gsrc into LDS at lds_off.
// D# built per CDNA5 ISA 8.3/8.4:
//   group0: [1:0]=count=1 | [63:32]=lds_addr | [120:64]=global_addr | [127:126]=type=2
//   group1: [17:16]=data_size=2 (4B) | tensor_dim0=Cc | tensor_dim1=WIN |
//           tile_dim0=Cc | tile_dim1=WIN | tile_dim2=0 | tensor_dim0_stride=Cc
//   groups 2/3: zero (2-D tensor; tile_dim3/4 unused)
// Tile == tensor window, so no OOB path is exercised.
__device__ __forceinline__ void tdm_load_rows(const float* gsrc, unsigned lds_off) {
    unsigned long long ga = (unsigned long long)(uintptr_t)gsrc;
    v4u g0;
    g0.x = 1u;                                                  // count=1, user mode
    g0.y = lds_off;                                             // lds_addr[31:0]
    g0.z = (unsigned)(ga & 0xffffffffull);                      // global_addr[31:0]
    g0.w = (unsigned)((ga >> 32) & 0x01ffffffull) | (2u << 30); // ga[56:32] | type=2
    v8i g1;
    g1[0] = (int)(2u << 16);                                    // wg_mask=0, data_size=4B
    g1[1] = (int)((Cc & 0xffff) << 16);                         // tensor_dim0 lo16
    g1[2] = (int)(((Cc >> 16) & 0xffff) | ((WIN & 0xffff) << 16)); // td0 hi16 | td1 lo16
    g1[3] = (int)(((WIN >> 16) & 0xffff) | ((Cc & 0xffff) << 16)); // td1 hi16 | tile_dim0
    g1[4] = (int)(WIN & 0xffff);                                // tile_dim1=WIN, tile_dim2=0
    g1[5] = (int)Cc;                                            // tensor_dim0_stride lo32
    g1[6] = 0;                                                  // stride0 hi16 | stride1 lo16
    g1[7] = 0;                                                  // stride1 hi32
    v4i z4 = {0, 0, 0, 0};
#if defined(__clang_major__) && (__clang_major__ >= 23)
    v8i z8 = {0, 0, 0, 0, 0, 0, 0, 0};
    __builtin_amdgcn_tensor_load_to_lds(g0, g1, z4, z4, z8, 0);
#else
    __builtin_amdgcn_tensor_load_to_lds(g0, g1, z4, z4, 0);
#endif
}

// One wave32 per batch element. Lane l owns states s = 5l .. 5l+4.
// States >= 129 use BLANK emissions; they only consume lower states, never
// feed back into valid states, so they are inert.
__global__ __launch_bounds__(32) void ctc_fwd_tdm(const int*  __restrict__ labels,
                                                  const float* __restrict__ probs,
                                                  float* __restrict__ out) {
    __shared__ float tb[2][WIN * Cc];            // 2 x 16KB double buffer

    const int b    = blockIdx.x;
    const int lane = threadIdx.x;                // wave32: lane == threadIdx.x
    const int*   lab = labels + b * Uu;
    const float* rp  = probs + (size_t)b * Tt * Cc;

    // Per-state class index e[j] and skip-transition flag (uniform over t).
    int  e[5];
    bool skip[5];
#pragma unroll
    for (int j = 0; j < 5; ++j) {
        int s = 5 * lane + j;
        if (s < Ss && (s & 1)) {
            int u = (s - 1) >> 1;
            e[j]    = lab[u];
            skip[j] = (u == 0) || (e[j] != lab[u - 1]);  // u==0 harmless: halo is NEG
        } else {
            e[j]    = BLANKC;
            skip[j] = false;
        }
    }

    float a[5];                                  // alpha, initialized at t==0 below

    // Prime the DMA pipeline: window 0 in flight before the loop.
    tdm_load_rows(rp, (unsigned)(uintptr_t)&tb[0][0]);

    for (int w = 0; w < NW; ++w) {
        // Overlap: issue window w+1 into the other buffer, then wait until only
        // it may be outstanding => window w (in-order retirement) is resident.
        if (w + 1 < NW) {
            tdm_load_rows(rp + (size_t)(w + 1) * WIN * Cc,
                          (unsigned)(uintptr_t)&tb[(w + 1) & 1][0]);
            __builtin_amdgcn_s_wait_tensorcnt((short)1);
        } else {
            __builtin_amdgcn_s_wait_tensorcnt((short)0);
        }
        const float* buf = &tb[w & 1][0];

#pragma unroll
        for (int i = 0; i < WIN; ++i) {
            const int t = w * WIN + i;
            // Gather this step's 5 emission probs from LDS (ds_load_b32;
            // the whole window is resident, so the scheduler may hoist these).
            float p0 = buf[i * Cc + e[0]];
            float p1 = buf[i * Cc + e[1]];
            float p2 = buf[i * Cc + e[2]];
            float p3 = buf[i * Cc + e[3]];
            float p4 = buf[i * Cc + e[4]];
            float lp0 = __logf(p0 + EPSV);
            float lp1 = __logf(p1 + EPSV);
            float lp2 = __logf(p2 + EPSV);
            float lp3 = __logf(p3 + EPSV);
            float lp4 = __logf(p4 + EPSV);
            if (t == 0) {
                // alpha0: states 0 and 1 start at emission log-prob, rest NEG.
                a[0] = (5 * lane + 0 < 2) ? lp0 : NEGV;
                a[1] = (5 * lane + 1 < 2) ? lp1 : NEGV;
                a[2] = NEGV; a[3] = NEGV; a[4] = NEGV;
            } else {
                // Halo: previous lane's top two OLD alphas = alpha[5l-1], alpha[5l-2].
                float h1 = __shfl_up(a[4], 1, 32);
                float h2 = __shfl_up(a[3], 1, 32);
                if (lane == 0) { h1 = NEGV; h2 = NEGV; }   // alpha[-1], alpha[-2]
                // 5 independent updates (ILP); all read OLD alphas.
                float n0 = lae3(a[0], h1,   skip[0] ? h2   : NEGV) + lp0;
                float n1 = lae3(a[1], a[0], skip[1] ? h1   : NEGV) + lp1;
                float n2 = lae3(a[2], a[1], skip[2] ? a[0] : NEGV) + lp2;
                float n3 = lae3(a[3], a[2], skip[3] ? a[1] : NEGV) + lp3;
                float n4 = lae3(a[4], a[3], skip[4] ? a[2] : NEGV) + lp4;
                a[0] = n0; a[1] = n1; a[2] = n2; a[3] = n3; a[4] = n4;
            }
        }
    }

    // loss = -logaddexp(alpha[S-1], alpha[S-2]) = -lae2(state128, state127).
    // Lane 25 owns states 125..129: j=3 -> 128, j=2 -> 127.
    if (lane == 25)
        out[b] = -lae2(a[3], a[2]);
}

extern "C" void kernel_launch(void* const* d_in, const int* in_sizes, int n_in,
                              void* d_out, int out_size, void* d_ws, size_t ws_size,
                              hipStream_t stream) {
    (void)in_sizes; (void)n_in; (void)d_ws; (void)ws_size; (void)out_size;
    const int*   y_true = (const int*)d_in[0];   // [B, U] int32
    const float* y_pred = (const float*)d_in[1]; // [B, T, C] float32
    float*       out    = (float*)d_out;         // [B, 1] float32

    ctc_fwd_tdm<<<dim3(Bsz), dim3(32), 0, stream>>>(y_true, y_pred, out);
}